// DiffeqSolver_KL_80599356276799
// MI455X (gfx1250) — compile-verified
//
#include <hip/hip_runtime.h>
#include <cmath>

// Elementwise, memory-bound: out = EPS * b' * ((s + log(C-b') - log(s'+C))*S1 - S2)
// S1/S2 are scalar time-sums computed on host. Input d_in[0] (b_phi_zt) is unused.
// CDNA5 path: double-buffered global_load_async_to_lds_b128 (+NT hint) with
// s_wait_asynccnt, ds_load_b128 readback, nontemporal b128 stores.

typedef __attribute__((ext_vector_type(4))) float v4f;

#define TPB 256
#define EPS_F 0.001f
#define C_F 0.01f

// Issue 3 async 16B loads (one per input stream) into per-lane LDS slots of
// buffer `buf`. Single static __shared__ array => LDS base offset 0, so the
// slot byte address is computed directly. ASYNCcnt += 3 per call (per wave).
// NT hint: pure streaming data, no reuse -> don't thrash WGP$/L2.
__device__ __forceinline__ void issue_async3(const float* bd, const float* sp,
                                             const float* sd, unsigned buf,
                                             long long i4, unsigned lane_slot) {
  unsigned goff = (unsigned)(i4 << 4);  // byte offset into each stream (<2^31)
  unsigned l0 = ((buf * 3u + 0u) * TPB + lane_slot) << 4;
  unsigned l1 = ((buf * 3u + 1u) * TPB + lane_slot) << 4;
  unsigned l2 = ((buf * 3u + 2u) * TPB + lane_slot) << 4;
  asm volatile("global_load_async_to_lds_b128 %0, %1, %2 th:TH_LOAD_NT"
               :: "v"(l0), "v"(goff), "s"(bd) : "memory");
  asm volatile("global_load_async_to_lds_b128 %0, %1, %2 th:TH_LOAD_NT"
               :: "v"(l1), "v"(goff), "s"(sp) : "memory");
  asm volatile("global_load_async_to_lds_b128 %0, %1, %2 th:TH_LOAD_NT"
               :: "v"(l2), "v"(goff), "s"(sd) : "memory");
}

__global__ __launch_bounds__(TPB) void diffeq_kl_kernel(
    const float* __restrict__ bd, const float* __restrict__ sp,
    const float* __restrict__ sd, float* __restrict__ out,
    long long n4, long long n, float S1, float S2) {
  // [double-buffer][stream][lane] : 24 KB of the 320 KB WGP LDS
  __shared__ v4f smem[2][3][TPB];

  // CRITICAL: let smem's address escape into asm with a memory clobber.
  // Without this, alias analysis proves the async-LDS asm writes (which only
  // receive integer LDS addresses) cannot touch smem, folds the readbacks of
  // a "never-written" array to undef, and deletes every ds_load.
  asm volatile("" ::"v"((const void*)&smem[0][0][0]) : "memory");

  const long long stride4 = (long long)gridDim.x * TPB;
  const long long tid4 = (long long)blockIdx.x * TPB + threadIdx.x;
  v4f* __restrict__ out4 = (v4f*)out;

  // Prologue: stage tile 0 into buffer 0.
  if (tid4 < n4) issue_async3(bd, sp, sd, 0u, tid4, threadIdx.x);

  unsigned cur = 0u;
  for (long long i = tid4; i < n4; i += stride4) {
    long long inx = i + stride4;
    if (inx < n4) {
      // Stage next tile into the other buffer, then wait until only those
      // 3 newest async ops remain outstanding => current tile is in LDS.
      issue_async3(bd, sp, sd, cur ^ 1u, inx, threadIdx.x);
      asm volatile("s_wait_asynccnt 0x3" ::: "memory");
    } else {
      asm volatile("s_wait_asynccnt 0x0" ::: "memory");
    }

    v4f b = smem[cur][0][threadIdx.x];
    v4f s = smem[cur][1][threadIdx.x];
    v4f d = smem[cur][2][threadIdx.x];
    v4f r;
#pragma unroll
    for (int c = 0; c < 4; ++c) {
      float K = s[c] + __logf(C_F - b[c]) - __logf(d[c] + C_F);
      r[c] = EPS_F * b[c] * (K * S1 - S2);
    }
    __builtin_nontemporal_store(r, &out4[i]);
    cur ^= 1u;
  }

  // Tail: n % 4 leftover scalars (n is 2^25 here so this is normally empty).
  long long te = (n4 << 2) + (long long)threadIdx.x;
  if (blockIdx.x == 0 && te < n) {
    float b = bd[te];
    float K = sp[te] + __logf(C_F - b) - __logf(sd[te] + C_F);
    out[te] = EPS_F * b * (K * S1 - S2);
  }
}

extern "C" void kernel_launch(void* const* d_in, const int* in_sizes, int n_in,
                              void* d_out, int out_size, void* d_ws, size_t ws_size,
                              hipStream_t stream) {
  (void)n_in; (void)out_size; (void)d_ws; (void)ws_size;
  // setup_inputs order: [0]=b_phi_zt (UNUSED), [1]=b_phi_zt_deriv,
  //                     [2]=s_phi_zt, [3]=s_phi_zt_deriv
  const float* bd = (const float*)d_in[1];
  const float* sp = (const float*)d_in[2];
  const float* sd = (const float*)d_in[3];
  float* out = (float*)d_out;

  long long n = (long long)in_sizes[1];
  long long n4 = n >> 2;

  // Host-side scalar time-sums (T=1000, t=1..998), double precision.
  double s1 = 0.0, s2 = 0.0;
  for (int t = 1; t <= 998; ++t) {
    double m = -1.0 + 0.001 * (double)t;
    double lm = log(-m);           // log(-m), m in (-1,0)
    double a = -1.0 / (m * lm);
    double c = log(-lm);
    s1 += a;
    s2 += a * c;
  }

  long long nb = (n4 + TPB - 1) / TPB;
  if (nb < 1) nb = 1;
  int blocks = (int)(nb < 4096 ? nb : 4096);

  diffeq_kl_kernel<<<blocks, TPB, 0, stream>>>(bd, sp, sd, out, n4, n,
                                               (float)s1, (float)s2);
}